// TorchSolver_22823456211151
// MI455X (gfx1250) — compile-verified
//
#include <hip/hip_runtime.h>

#ifndef __has_builtin
#define __has_builtin(x) 0
#endif

#define LOG2E 1.4426950408889634f
#define NGENOME 1000
#define NSAMP   32
#define TABLE_F32 (NGENOME * NSAMP)       // 32000 floats = 128 KB per table
#define TABLE_V4  (TABLE_F32 / 4)         // 8000 int4 chunks

typedef float f32x4 __attribute__((ext_vector_type(4)));
typedef int   v4i   __attribute__((ext_vector_type(4)));
typedef __attribute__((address_space(1))) v4i* gv4i_ptr;
typedef __attribute__((address_space(3))) v4i* lv4i_ptr;

__global__ void zero_f32_kernel(float* __restrict__ p, int n) {
    int i = blockIdx.x * blockDim.x + threadIdx.x;
    int stride = gridDim.x * blockDim.x;
    for (; i < n; i += stride) p[i] = 0.0f;
}

__device__ __forceinline__ float fast_exp_term(float a, float b, float p) {
    // exp(a + 1 - p*b) = exp2(((a - p*b) + 1) * log2e)
    const float t = __builtin_fmaf(-p, b, a);
    const float u = __builtin_fmaf(t, LOG2E, LOG2E);
#if __has_builtin(__builtin_amdgcn_exp2f)
    return __builtin_amdgcn_exp2f(u);
#else
    return exp2f(u);
#endif
}

__launch_bounds__(1024, 1)
__global__ void ptr_segsum_kernel(const float* __restrict__ A,
                                  const float* __restrict__ B,
                                  const float* __restrict__ pos,
                                  const int*   __restrict__ gidx,
                                  const int*   __restrict__ sidx,
                                  float*       __restrict__ out,
                                  int M) {
    // Entire A and B tables staged in LDS: 2 x 128 KB of the 320 KB WGP LDS.
    __shared__ __attribute__((aligned(16))) float ldsA[TABLE_F32];
    __shared__ __attribute__((aligned(16))) float ldsB[TABLE_F32];

    const int tid = threadIdx.x;

    // ---- Stage A/B tables into LDS with gfx1250 async copies (ASYNCcnt) ----
#if __has_builtin(__builtin_amdgcn_global_load_async_to_lds_b128)
    {
        gv4i_ptr gA = (gv4i_ptr)(const_cast<float*>(A));
        gv4i_ptr gB = (gv4i_ptr)(const_cast<float*>(B));
        lv4i_ptr lA = (lv4i_ptr)ldsA;
        lv4i_ptr lB = (lv4i_ptr)ldsB;
        for (int i = tid; i < TABLE_V4; i += 1024) {
            __builtin_amdgcn_global_load_async_to_lds_b128(gA + i, lA + i, 0, 0);
            __builtin_amdgcn_global_load_async_to_lds_b128(gB + i, lB + i, 0, 0);
        }
#if __has_builtin(__builtin_amdgcn_s_wait_asynccnt)
        __builtin_amdgcn_s_wait_asynccnt(0);
#else
        asm volatile("s_wait_asynccnt 0" ::: "memory");
#endif
    }
#else
    for (int i = tid; i < TABLE_V4; i += 1024) {
        ((f32x4*)ldsA)[i] = ((const f32x4*)A)[i];
        ((f32x4*)ldsB)[i] = ((const f32x4*)B)[i];
    }
#endif
    __syncthreads();

    // ---- One wave per 32-gene chunk; lane = sample index ----
    const int lane = tid & 31;
    const int w    = (blockIdx.x * blockDim.x + tid) >> 5;
    const int nw   = (gridDim.x * blockDim.x) >> 5;
    const int C    = M >> 5;               // full 32-gene chunks

    for (int c = w; c < C; c += nw) {
        const int base = c << 5;
        // Coalesced metadata fetch: lane l holds gene (base+l)'s scalars.
        const float pv = pos[base + lane];
        const int   gv = gidx[base + lane];
        const int   qv = sidx[base + lane];

#pragma unroll 8
        for (int j = 0; j < 32; ++j) {
            // Broadcast gene j's metadata to all lanes: v_readlane, no memory.
            const float p = __builtin_bit_cast(float,
                __builtin_amdgcn_readlane(__builtin_bit_cast(int, pv), j));
            const int   g = __builtin_amdgcn_readlane(gv, j);
            const int   q = __builtin_amdgcn_readlane(qv, j);

            const float a = ldsA[(g << 5) + lane];   // conflict-free row read
            const float b = ldsB[(g << 5) + lane];
            const float v = fast_exp_term(a, b, p);

            // 12.8 MB accumulator lives in 192 MB L2; native f32 atomic, no return
            unsafeAtomicAdd(&out[(q << 5) + lane], v);
        }
    }

    // ---- Tail genes (M % 32), handled by global wave 0 only ----
    if (w == 0) {
        for (int m = C << 5; m < M; ++m) {
            const int mm = __builtin_amdgcn_readfirstlane(m);
            const float p = pos[mm];
            const int   g = gidx[mm];
            const int   q = sidx[mm];
            const float a = ldsA[(g << 5) + lane];
            const float b = ldsB[(g << 5) + lane];
            const float v = fast_exp_term(a, b, p);
            unsafeAtomicAdd(&out[(q << 5) + lane], v);
        }
    }
}

extern "C" void kernel_launch(void* const* d_in, const int* in_sizes, int n_in,
                              void* d_out, int out_size, void* d_ws, size_t ws_size,
                              hipStream_t stream) {
    const float* A    = (const float*)d_in[0];   // [N, S]
    const float* B    = (const float*)d_in[1];   // [N, S]
    const float* pos  = (const float*)d_in[2];   // [M]
    const int*   gidx = (const int*)  d_in[3];   // [M]
    const int*   sidx = (const int*)  d_in[4];   // [M]
    float*       out  = (float*)d_out;           // [K, S] flattened
    const int M = in_sizes[2];

    // Zero the accumulator every launch (atomics accumulate into it).
    {
        int threads = 256;
        int blocks  = (out_size + threads - 1) / threads;
        if (blocks > 4096) blocks = 4096;
        zero_f32_kernel<<<blocks, threads, 0, stream>>>(out, out_size);
    }

    // Main kernel: 512 workgroups x 1024 threads (32 waves each, 1 WG per WGP).
    ptr_segsum_kernel<<<512, 1024, 0, stream>>>(A, B, pos, gidx, sidx, out, M);
}